// MaskedSelfAttention_20323785245297
// MI455X (gfx1250) — compile-verified
//
#include <hip/hip_runtime.h>

typedef __attribute__((ext_vector_type(2))) float v2f;
typedef __attribute__((ext_vector_type(8))) float v8f;

#define B_  4
#define S_  4096
#define E_  128
#define KD_ 64
#define TAILN 4032.0f   // S - 64 zero-score columns per softmax row

// ---------------- workspace layout (floats) ----------------
// Km   : [B][64][64]   masked K (Km[t][d] = key[t][d] if d>=t else 0)   @ 0      (16384)
// V64t : [B][128][64]  V rows 0..63, transposed (v-major, t contiguous) @ 16384  (32768)
// Ct   : [B][64][128]  C^T  (t-major, e contiguous); scores = x @ C     @ 49152  (32768)
// Vsum : [B][128]      sum_{t>=64} V[t][:]                              @ 81920  (512)
// part : [B][63][128]  partial tail sums of x                           @ 82432  (32256)

// ---- prepass 1: partial tail sums of x over t in [64,4096), 64 rows/block ----
__global__ __launch_bounds__(128) void k_partial(const float* __restrict__ x,
                                                 float* __restrict__ part) {
    int b = blockIdx.x / 63, c = blockIdx.x % 63;
    int e = threadIdx.x;
    const float* p = x + ((size_t)b * S_ + 64 + c * 64) * E_ + e;
    float s = 0.f;
    #pragma unroll 8
    for (int r = 0; r < 64; ++r) s += p[(size_t)r * E_];
    part[(size_t)blockIdx.x * E_ + e] = s;
}

// ---- prepass 2: Vsum[b][v] = (sum_tail x[b]) . Wv[v] ----
__global__ __launch_bounds__(128) void k_vsum(const float* __restrict__ part,
                                              const float* __restrict__ Wv,
                                              float* __restrict__ Vsum) {
    __shared__ float xs[E_];
    int b = blockIdx.x, t = threadIdx.x;
    float s = 0.f;
    for (int c = 0; c < 63; ++c) s += part[((size_t)b * 63 + c) * E_ + t];
    xs[t] = s;
    __syncthreads();
    const float* wr = Wv + (size_t)t * E_;
    float d = 0.f;
    #pragma unroll 8
    for (int e = 0; e < E_; ++e) d += xs[e] * wr[e];
    Vsum[b * E_ + t] = d;
}

// ---- prepass 3: masked Km[b][t][d] and V64t[b][v][t] (first 64 rows only) ----
__global__ __launch_bounds__(256) void k_prepkv(const float* __restrict__ x,
                                                const float* __restrict__ Wk,
                                                const float* __restrict__ Wv,
                                                float* __restrict__ Km,
                                                float* __restrict__ V64t) {
    int gid = blockIdx.x * 256 + threadIdx.x;
    if (gid < B_ * 64 * 64) {                 // Km element
        int b = gid / 4096, rem = gid % 4096, t = rem / 64, d = rem % 64;
        float s = 0.f;
        if (d >= t) {                          // tril(K^T): keep iff t <= d
            const float* xr = x + ((size_t)b * S_ + t) * E_;
            const float* wr = Wk + (size_t)d * E_;
            #pragma unroll 8
            for (int e = 0; e < E_; ++e) s += xr[e] * wr[e];
        }
        Km[gid] = s;
    } else {                                   // V64t element (v-major, t contiguous)
        int g = gid - B_ * 64 * 64;
        int b = g / 8192, rem = g % 8192, v = rem / 64, t = rem % 64;
        const float* xr = x + ((size_t)b * S_ + t) * E_;
        const float* wr = Wv + (size_t)v * E_;
        float s = 0.f;
        #pragma unroll 8
        for (int e = 0; e < E_; ++e) s += xr[e] * wr[e];
        V64t[g] = s;
    }
}

// ---- prepass 4: Ct[b][t][e] = sum_d Wq[d][e] * Km[b][t][d]  (scores = x @ C) ----
__global__ __launch_bounds__(256) void k_prepc(const float* __restrict__ Wq,
                                               const float* __restrict__ Km,
                                               float* __restrict__ Ct) {
    int gid = blockIdx.x * 256 + threadIdx.x;   // < 32768
    int b = gid / 8192, rem = gid % 8192, t = rem / 128, e = rem % 128;
    const float* kr = Km + ((size_t)b * 64 + t) * 64;
    float s = 0.f;
    #pragma unroll 8
    for (int d = 0; d < 64; ++d) s += Wq[d * E_ + e] * kr[d];
    Ct[gid] = s;
}

// ---- main fused kernel: 4 waves/block, 16 rows/wave, fp32 WMMA 16x16x4 ----
__global__ __launch_bounds__(128) void attn_main(const float* __restrict__ x,
                                                 const float* __restrict__ Ct,
                                                 const float* __restrict__ V64t,
                                                 const float* __restrict__ Vsum,
                                                 float* __restrict__ out) {
    __shared__ float ldsS[4][16][64];   // per-wave 16x64 scores / probs
    __shared__ float rowTail[4][16];    // exp(-m) per row
    __shared__ float rowInv[4][16];     // 1/Z per row

    const int tid  = threadIdx.x;
    const int w    = tid >> 5;
    const int lane = tid & 31;
    const int hi   = lane >> 4;   // lane half selects K pair {0,1} vs {2,3}
    const int lm   = lane & 15;   // A: row M ; B: column N
    const int b    = blockIdx.x >> 6;
    const int rb   = blockIdx.x & 63;
    const int i0   = rb * 64 + w * 16;

    // ---- phase 1: scores(16x64) = x_tile(16x128) @ C_b(128x64) ----
    const float* xrow = x + ((size_t)(b * S_ + i0 + lm)) * E_ + 2 * hi;
    const float* ctb  = Ct + (size_t)b * 64 * 128;
    v8f acc[4] = {};
    #pragma unroll 4
    for (int e0 = 0; e0 < 128; e0 += 4) {
        v2f a = *(const v2f*)(xrow + e0);                       // A[m][e0+2hi..+1]
        #pragma unroll
        for (int nc = 0; nc < 4; ++nc) {
            v2f bb = *(const v2f*)(ctb + (size_t)(nc * 16 + lm) * 128 + e0 + 2 * hi);
            acc[nc] = __builtin_amdgcn_wmma_f32_16x16x4_f32(
                false, a, false, bb, (short)0, acc[nc], false, false);
        }
    }
    #pragma unroll
    for (int nc = 0; nc < 4; ++nc)
        #pragma unroll
        for (int r = 0; r < 8; ++r)
            ldsS[w][r + 8 * hi][nc * 16 + lm] = acc[nc][r];     // D layout -> row major
    __syncthreads();

    // ---- phase 2: softmax factors (64 live cols + 4032 implicit zeros) ----
    if (lane < 16) {
        float m = 0.0f;                                          // zeros floor the max
        #pragma unroll 8
        for (int t = 0; t < 64; ++t) m = fmaxf(m, ldsS[w][lane][t]);
        float eT = __expf(-m);
        float Z  = TAILN * eT;
        #pragma unroll 8
        for (int t = 0; t < 64; ++t) {
            float p = __expf(ldsS[w][lane][t] - m);
            Z += p;
            ldsS[w][lane][t] = p;                                // un-normalized P
        }
        rowTail[w][lane] = eT;
        rowInv[w][lane]  = 1.0f / Z;
    }
    __syncthreads();

    // ---- phase 3: out(16x128) = P(16x64) @ V64 + tail, scaled by 1/Z ----
    const float* vtb = V64t + (size_t)b * 128 * 64;
    v8f oacc[8] = {};
    #pragma unroll 4
    for (int t0 = 0; t0 < 64; t0 += 4) {
        v2f a = *(const v2f*)(&ldsS[w][lm][t0 + 2 * hi]);
        #pragma unroll
        for (int nc = 0; nc < 8; ++nc) {
            v2f bb = *(const v2f*)(vtb + (size_t)(nc * 16 + lm) * 64 + t0 + 2 * hi);
            oacc[nc] = __builtin_amdgcn_wmma_f32_16x16x4_f32(
                false, a, false, bb, (short)0, oacc[nc], false, false);
        }
    }
    float* ob = out + (size_t)(b * S_ + i0) * E_;
    #pragma unroll
    for (int nc = 0; nc < 8; ++nc) {
        float vs = Vsum[b * E_ + nc * 16 + lm];
        #pragma unroll
        for (int r = 0; r < 8; ++r) {
            int row = r + 8 * hi;
            float val = (oacc[nc][r] + rowTail[w][row] * vs) * rowInv[w][row];
            ob[(size_t)row * E_ + nc * 16 + lm] = val;
        }
    }
}

extern "C" void kernel_launch(void* const* d_in, const int* in_sizes, int n_in,
                              void* d_out, int out_size, void* d_ws, size_t ws_size,
                              hipStream_t stream) {
    const float* x  = (const float*)d_in[0];
    const float* Wk = (const float*)d_in[1];
    const float* Wq = (const float*)d_in[2];
    const float* Wv = (const float*)d_in[3];
    float* out = (float*)d_out;
    float* ws  = (float*)d_ws;

    float* Km   = ws;            // 16384
    float* V64t = ws + 16384;    // 32768
    float* Ct   = ws + 49152;    // 32768
    float* Vsum = ws + 81920;    // 512
    float* part = ws + 82432;    // 32256

    k_partial<<<B_ * 63, 128, 0, stream>>>(x, part);
    k_vsum  <<<B_,       128, 0, stream>>>(part, Wv, Vsum);
    k_prepkv<<<192,      256, 0, stream>>>(x, Wk, Wv, Km, V64t);
    k_prepc <<<128,      256, 0, stream>>>(Wq, Km, Ct);
    attn_main<<<256,     128, 0, stream>>>(x, Ct, V64t, Vsum, out);
}